// fab_penalty_ls_curve_39178691674437
// MI455X (gfx1250) — compile-verified
//
#include <hip/hip_runtime.h>
#include <stdint.h>
#include <math.h>

typedef __attribute__((ext_vector_type(2))) float v2f;
typedef __attribute__((ext_vector_type(8))) float v8f;

#define TX 64                  // tile cols (axis 1, contiguous)
#define TY 32                  // tile rows (axis 0)
#define SMW 72                 // lds cols: global cols [baseC-4, baseC+68)  (16B aligned rows)
#define SMH (TY + 4)           // 36 rows: global rows [baseR-2, baseR+TY+2)
#define BLOCK 256

__device__ __forceinline__ int clampi(int v, int lo, int hi) {
  return v < lo ? lo : (v > hi ? hi : v);
}

__global__ __launch_bounds__(BLOCK)
void curve_penalty_kernel(const float* __restrict__ eps,
                          const float* __restrict__ gsz,
                          float* __restrict__ ws,
                          float* __restrict__ out,
                          int N, int mode) {
  __shared__ float tile[SMH * SMW];
  __shared__ float red[BLOCK];

  const int baseR = blockIdx.y * TY;
  const int baseC = blockIdx.x * TX;

  const bool colEdge = (baseC < 4) || (baseC + 68 > N);
  const bool rowEdge = (baseR < 2) || (baseR > N - TY - 2);

  // ---- CDNA5 async fill (global -> LDS, ASYNCcnt). b128 fast path, b32 clamped edge path ----
  if (!colEdge) {
    // 36 rows x 18 float4-chunks; rows clamped, columns in-bounds & 16B aligned
    for (int idx = threadIdx.x; idx < SMH * 18; idx += BLOCK) {
      int tr = idx / 18;
      int ch = idx - tr * 18;
      int gr = clampi(baseR + tr - 2, 0, N - 1);
      uint64_t ga = (uint64_t)(uintptr_t)(eps + (size_t)gr * (size_t)N
                                              + (size_t)(baseC - 4 + ch * 4));
      uint32_t la = (uint32_t)(uintptr_t)(&tile[tr * SMW + ch * 4]);
      asm volatile("global_load_async_to_lds_b128 %0, %1, off"
                   :: "v"(la), "v"(ga) : "memory");
    }
  } else {
    for (int idx = threadIdx.x; idx < SMH * SMW; idx += BLOCK) {
      int tr = idx / SMW;
      int tc = idx - tr * SMW;
      int gr = clampi(baseR + tr - 2, 0, N - 1);
      int gc = clampi(baseC + tc - 4, 0, N - 1);
      uint64_t ga = (uint64_t)(uintptr_t)(eps + (size_t)gr * (size_t)N + (size_t)gc);
      uint32_t la = (uint32_t)(uintptr_t)(&tile[idx]);
      asm volatile("global_load_async_to_lds_b32 %0, %1, off"
                   :: "v"(la), "v"(ga) : "memory");
    }
  }
  asm volatile("s_wait_asynccnt 0" ::: "memory");
  __syncthreads();

  const float d     = gsz[0];
  const float invd  = 1.0f / d;
  const float inv2d = 0.5f * invd;
  const float dd    = d * d;
  const float SC    = 1e-12f;
  const float PI_D  = 2.8559933214452666f;   // pi / (1.1 * MIN_FEATURE_SIZE)
  const float EVF   = 4.641588833612779e-6f; // (1e-32)^(1/6)

  // thread ownership: row r, 8 consecutive cols starting at seg
  const int rr  = (int)(threadIdx.x >> 3);          // 0..31
  const int seg = (int)(threadIdx.x & 7) * 8;       // 0,8,..,56
  const int r   = baseR + rr;
  const int lr  = rr + 2;                            // lds row of r

  float acc = 0.0f;

  if (!colEdge && !rowEdge) {
    // ------- branch-free interior path: register-blocked 5-row footprint -------
    float rc[16], ru[16], rd[16], ruu[8], rdd[8];
    const int bC  = lr * SMW + seg;        // tile[lr][seg+0] == global col baseC+seg-4
#pragma unroll
    for (int q = 0; q < 4; ++q) {
      *(float4*)&rc[q * 4] = *(const float4*)&tile[bC + q * 4];
      *(float4*)&ru[q * 4] = *(const float4*)&tile[bC - SMW + q * 4];
      *(float4*)&rd[q * 4] = *(const float4*)&tile[bC + SMW + q * 4];
    }
#pragma unroll
    for (int q = 0; q < 2; ++q) {
      *(float4*)&ruu[q * 4] = *(const float4*)&tile[bC - 2 * SMW + 4 + q * 4];
      *(float4*)&rdd[q * 4] = *(const float4*)&tile[bC + 2 * SMW + 4 + q * 4];
    }
    // rc[k] == phi[r][baseC+seg-4+k]; output col j maps to index i=j+4; ruu/rdd[j] == phi[r-+2][col j]
#pragma unroll
    for (int j = 0; j < 8; ++j) {
      const int i = j + 4;
      float ex  = (rd[i] - ru[i]) * inv2d + SC;         // gx(r,c)
      float ey  = (rc[i + 1] - rc[i - 1]) * inv2d + SC; // gy(r,c)
      float gxp = (rdd[j] - rc[i]) * inv2d;             // gx(r+1,c)
      float gxm = (rc[i] - ruu[j]) * inv2d;             // gx(r-1,c)
      float exx = (gxp - gxm) * inv2d;
      float gxr = (rd[i + 1] - ru[i + 1]) * inv2d;      // gx(r,c+1)
      float gxl = (rd[i - 1] - ru[i - 1]) * inv2d;      // gx(r,c-1)
      float exy = (gxr - gxl) * inv2d;
      float gyr = (rc[i + 2] - rc[i]) * inv2d;          // gy(r,c+1)
      float gyl = (rc[i] - rc[i - 2]) * inv2d;          // gy(r,c-1)
      float eyy = (gyr - gyl) * inv2d;
      float ev  = fmaxf(sqrtf(ex * ex + ey * ey), EVF);
      float k   = (ex * ex * eyy - 2.0f * ex * ey * exy + ey * ey * exx) / (ev * ev * ev);
      float cc  = fabsf(k * atanf(ev / (rc[i] + 1e-6f))) - PI_D;
      float v   = fmaxf(cc, 0.0f) * dd;
      if (v == v) acc += v;                             // nansum: drop NaNs
    }
  } else {
    // ------- generic border path: np.gradient edge_order=1 one-sided formulas -------
    auto P = [&](int tr, int tc) -> float { return tile[tr * SMW + tc]; };
    auto GX = [&](int rA, int cA) -> float {
      int tr = rA - baseR + 2, tc = cA - baseC + 4;
      if (rA == 0)     return (P(tr + 1, tc) - P(tr, tc)) * invd;
      if (rA == N - 1) return (P(tr, tc) - P(tr - 1, tc)) * invd;
      return (P(tr + 1, tc) - P(tr - 1, tc)) * inv2d;
    };
    auto GY = [&](int rA, int cA) -> float {
      int tr = rA - baseR + 2, tc = cA - baseC + 4;
      if (cA == 0)     return (P(tr, tc + 1) - P(tr, tc)) * invd;
      if (cA == N - 1) return (P(tr, tc) - P(tr, tc - 1)) * invd;
      return (P(tr, tc + 1) - P(tr, tc - 1)) * inv2d;
    };
    for (int j = 0; j < 8; ++j) {
      const int c = baseC + seg + j;
      if (r >= N || c >= N) continue;
      float ex = GX(r, c) + SC;
      float ey = GY(r, c) + SC;
      float exx = (r == 0)     ? (GX(1, c) - GX(0, c)) * invd
                : (r == N - 1) ? (GX(N - 1, c) - GX(N - 2, c)) * invd
                               : (GX(r + 1, c) - GX(r - 1, c)) * inv2d;
      float exy = (c == 0)     ? (GX(r, 1) - GX(r, 0)) * invd
                : (c == N - 1) ? (GX(r, N - 1) - GX(r, N - 2)) * invd
                               : (GX(r, c + 1) - GX(r, c - 1)) * inv2d;
      float eyy = (c == 0)     ? (GY(r, 1) - GY(r, 0)) * invd
                : (c == N - 1) ? (GY(r, N - 1) - GY(r, N - 2)) * invd
                               : (GY(r, c + 1) - GY(r, c - 1)) * inv2d;
      float ev = fmaxf(sqrtf(ex * ex + ey * ey), EVF);
      float k  = (ex * ex * eyy - 2.0f * ex * ey * exy + ey * ey * exx) / (ev * ev * ev);
      float cc = fabsf(k * atanf(ev / (P(r - baseR + 2, c - baseC + 4) + 1e-6f))) - PI_D;
      float v  = fmaxf(cc, 0.0f) * dd;
      if (v == v) acc += v;
    }
  }

  red[threadIdx.x] = acc;
  __syncthreads();

  // ---- block reduction: 256 -> 32 via LDS, 32 -> 1 via V_WMMA_F32_16X16X4_F32 ----
  if (threadIdx.x < 32) {     // wave-uniform: wave 0 has full EXEC (WMMA requirement)
    float p = 0.0f;
#pragma unroll
    for (int i = 0; i < 8; ++i) p += red[threadIdx.x * 8 + i];
    // A(16x4): lane L<16 -> A[L][0]=p, lane L>=16 -> A[L-16][2]=p; B = ones(4x16)
    // => D[m][n] = p_m + p_{m+16}; lane0 holds D[0..7][0], lane16 holds D[8..15][0]
    v2f a; a.x = p;    a.y = 0.0f;
    v2f b; b.x = 1.0f; b.y = 1.0f;
    v8f cacc = {};
    cacc = __builtin_amdgcn_wmma_f32_16x16x4_f32(false, a, false, b,
                                                 (short)0, cacc, false, false);
    float s = cacc[0] + cacc[1] + cacc[2] + cacc[3]
            + cacc[4] + cacc[5] + cacc[6] + cacc[7];
    float s16 = __shfl(s, 16, 32);
    if (threadIdx.x == 0) {
      float total = s + s16;
      if (mode) ws[blockIdx.y * gridDim.x + blockIdx.x] = total;
      else      atomicAdd(out, total);
    }
  }
}

__global__ void zero_out_kernel(float* out) { out[0] = 0.0f; }

__global__ __launch_bounds__(256)
void reduce_partials_kernel(const float* __restrict__ ws,
                            float* __restrict__ out, int n) {
  __shared__ float red[256];
  float a = 0.0f;
  for (int i = threadIdx.x; i < n; i += 256) a += ws[i];
  red[threadIdx.x] = a;
  __syncthreads();
  for (int s = 128; s > 0; s >>= 1) {
    if ((int)threadIdx.x < s) red[threadIdx.x] += red[threadIdx.x + s];
    __syncthreads();
  }
  if (threadIdx.x == 0) out[0] = red[0];
}

extern "C" void kernel_launch(void* const* d_in, const int* in_sizes, int n_in,
                              void* d_out, int out_size, void* d_ws, size_t ws_size,
                              hipStream_t stream) {
  const float* eps = (const float*)d_in[0];
  const float* gsz = (const float*)d_in[1];
  float* out = (float*)d_out;

  int N = (int)(sqrt((double)in_sizes[0]) + 0.5);   // 8192 for the reference shape
  int gxt = (N + TX - 1) / TX;
  int gyt = (N + TY - 1) / TY;
  int nblk = gxt * gyt;
  bool twoStage = (ws_size >= (size_t)nblk * sizeof(float));

  if (!twoStage) zero_out_kernel<<<1, 1, 0, stream>>>(out);
  dim3 grid(gxt, gyt);
  curve_penalty_kernel<<<grid, BLOCK, 0, stream>>>(eps, gsz, (float*)d_ws, out,
                                                   N, twoStage ? 1 : 0);
  if (twoStage)
    reduce_partials_kernel<<<1, 256, 0, stream>>>((const float*)d_ws, out, nblk);
}